// QuantumLaserEngine_47098611368141
// MI455X (gfx1250) — compile-verified
//
#include <hip/hip_runtime.h>
#include <hip/hip_bf16.h>
#include <math.h>

// ---------------------------------------------------------------------------
// QuantumLaserEngine for MI455X (gfx1250, wave32).
// Pipeline (all on `stream`, graph-capture safe; float atomic sums are order-dependent):
//   k0  zero accumulators (emission, n_lasing, tension)
//   k1  pump GEMV via V_WMMA_F32_16X16X4_F32 -> lasing mask + n_lasing
//   k2  phase rotation: rot = states * exp(i*0.1*phase_vel)   (sincos)
//   k3  neighbor coupling: cos(dphi) rewritten as normalized dot product
//       (kills 126M cos evals; rot[] (67MB) lives in 192MB L2 for the gather)
//   k4  masked emission column-reduction -> em[1024]
//   k5  cavity_new + old cavity phase
//   k5b pred = [cav_new.re|cav_new.im] @ dec_W^T + dec_b  (wave-per-row dot)
//   k6  phase-lock + cavity feedback + per-row amp renorm + var partials
//   k7  tension = E[a^2]-E[a]^2
// Workspace use: ~134.3 MB (rot 64MB, newst 64MB, masks/accums small).
// ---------------------------------------------------------------------------

typedef __attribute__((ext_vector_type(2))) float v2f;
typedef __attribute__((ext_vector_type(8))) float v8f;

#define N_CELLS 16384
#define DIM     512
#define HIDDEN  512
#define KNB     15

// ---------------- K0: zero accumulators ----------------
__global__ void k0_init(float* __restrict__ em, int* __restrict__ nl,
                        float* __restrict__ tens) {
  int t = threadIdx.x;
  if (t < 2 * HIDDEN) em[t] = 0.0f;
  if (t == 0) { *nl = 0; tens[0] = 0.0f; tens[1] = 0.0f; }
}

// ---------------- K1: pump GEMV via f32 WMMA ----------------
// D = A(16x4) x B(4x16) + C, accumulated over K=512 in steps of 4.
// A[m,k] = x[d0+k] (replicated over m), B[k,n] = pump_W[cellBase+n, d0+k].
// f32 operand layout (ISA 7.12.2): lanes 0-15 carry K={0,1}, lanes 16-31
// carry K={2,3} -> one float2 load per lane per step for both A and B.
__global__ void k1_pump(const float* __restrict__ x,
                        const float* __restrict__ pump_W,
                        const float* __restrict__ pump_b,
                        const float* __restrict__ excited,
                        float* __restrict__ lasing,
                        int* __restrict__ nlasing) {
  const int lane = threadIdx.x & 31;
  const int waveGlobal = blockIdx.x * (blockDim.x >> 5) + (threadIdx.x >> 5);
  const int cellBase = waveGlobal * 16;
  const int n16  = lane & 15;
  const int hi2  = (lane >> 4) << 1;          // 0 (low half) or 2 (high half)
  const int cell = cellBase + n16;

  const float* wrow = pump_W + (size_t)cell * DIM;
  v8f c = {};
  for (int d0 = 0; d0 < DIM; d0 += 4) {
    v2f a = { x[d0 + hi2],    x[d0 + hi2 + 1]    };
    v2f b = { wrow[d0 + hi2], wrow[d0 + hi2 + 1] };
    c = __builtin_amdgcn_wmma_f32_16x16x4_f32(false, a, false, b,
                                              (short)0, c, false, false);
  }
  // D is replicated over M; VGPR0 of every lane holds pump-dot for cell n16.
  float t = c[0] + pump_b[cell];
  float p = 1.0f / (1.0f + __expf(-t));                 // sigmoid
  float exc = excited[cell] * 0.95f + p * 0.05f;
  exc = fminf(fmaxf(exc, 0.0f), 1.0f);
  float las = (exc > 0.5f) ? 1.0f : 0.0f;
  if (lane < 16) lasing[cell] = las;
  unsigned long long bal = __ballot((lane < 16) && (las > 0.5f));
  if (lane == 0) atomicAdd(nlasing, (int)__popcll(bal));
}

// ---------------- K2: free phase rotation ----------------
__global__ void k2_rotate(const float* __restrict__ cs_re,
                          const float* __restrict__ cs_im,
                          const float* __restrict__ pv,
                          float2* __restrict__ rot) {
  size_t idx = (size_t)blockIdx.x * blockDim.x + threadIdx.x;
  const size_t total = (size_t)N_CELLS * HIDDEN;
  const size_t stride = (size_t)gridDim.x * blockDim.x;
  for (; idx < total; idx += stride) {
    float re = cs_re[idx], im = cs_im[idx];
    float sn, cs;
    __sincosf(pv[idx] * 0.1f, &sn, &cs);
    rot[idx] = make_float2(re * cs - im * sn, re * sn + im * cs);
  }
}

// ---------------- K3: neighbor phase coupling ----------------
// cos(ang_i - ang_nb) * nb = ((s_i . s_nb) / (|s_i||s_nb|)) * s_nb
// new = 0.7*s + (0.3*0.1/15)*sum_k coup_k*nb_k = 0.7*s + 0.002*acc
__global__ void k3_couple(const float2* __restrict__ rot,
                          const int* __restrict__ neighbors,
                          float2* __restrict__ newst) {
  __shared__ int s_nb[KNB];
  const int i = blockIdx.x;
  if (threadIdx.x < KNB) {
    int nb = neighbors[i * KNB + threadIdx.x];
    s_nb[threadIdx.x] = nb;
    __builtin_prefetch(rot + (size_t)nb * HIDDEN, 0, 0);  // global_prefetch_b8
  }
  __syncthreads();
  for (int j = threadIdx.x; j < HIDDEN; j += blockDim.x) {
    float2 s = rot[(size_t)i * HIDDEN + j];
    float inv_i = __frsqrt_rn(s.x * s.x + s.y * s.y + 1e-30f);
    float accx = 0.0f, accy = 0.0f;
#pragma unroll
    for (int k = 0; k < KNB; ++k) {
      float2 t = rot[(size_t)s_nb[k] * HIDDEN + j];
      float invn = __frsqrt_rn(t.x * t.x + t.y * t.y + 1e-30f);
      float coup = (s.x * t.x + s.y * t.y) * inv_i * invn;
      accx = fmaf(coup, t.x, accx);
      accy = fmaf(coup, t.y, accy);
    }
    newst[(size_t)i * HIDDEN + j] =
        make_float2(0.7f * s.x + 0.002f * accx, 0.7f * s.y + 0.002f * accy);
  }
}

// ---------------- K4: masked emission column reduction ----------------
// grid(2,128): block covers 256 columns x 128 cells; coalesced float2 rows.
__global__ void k4_emission(const float2* __restrict__ newst,
                            const float* __restrict__ lasing,
                            float* __restrict__ em_re,
                            float* __restrict__ em_im) {
  const int j  = blockIdx.x * blockDim.x + threadIdx.x;   // 0..511
  const int i0 = blockIdx.y * 128;
  float sre = 0.0f, sim = 0.0f;
  for (int i = i0; i < i0 + 128; ++i) {
    if (lasing[i] > 0.0f) {                               // block-uniform branch
      float2 v = newst[(size_t)i * HIDDEN + j];
      sre += v.x; sim += v.y;
    }
  }
  atomicAdd(&em_re[j], sre);
  atomicAdd(&em_im[j], sim);
}

// ---------------- K5: cavity update + old cavity phase ----------------
__global__ void k5_cavity(const float* __restrict__ em_re,
                          const float* __restrict__ em_im,
                          const int* __restrict__ nlasing,
                          const float* __restrict__ cav_re,
                          const float* __restrict__ cav_im,
                          float* __restrict__ cnew_re,
                          float* __restrict__ cnew_im,
                          float* __restrict__ cphase) {
  const int j = threadIdx.x;            // blockDim == 512
  const int nl = *nlasing;
  const float inv_nl = 1.0f / (float)(nl > 1 ? nl : 1);
  float cr = cav_re[j], ci = cav_im[j];
  cnew_re[j] = (nl > 0) ? (0.8f * cr + 0.2f * em_re[j] * inv_nl) : cr;
  cnew_im[j] = (nl > 0) ? (0.8f * ci + 0.2f * em_im[j] * inv_nl) : ci;
  cphase[j]  = atan2f(ci, cr);          // phase of OLD cavity (per reference)
}

// ---------------- K5b: pred = out_real @ dec_W^T + dec_b ----------------
__global__ void k5b_pred(const float* __restrict__ cnew_re,
                         const float* __restrict__ cnew_im,
                         const float* __restrict__ dec_W,
                         const float* __restrict__ dec_b,
                         float* __restrict__ pred) {
  const int lane = threadIdx.x & 31;
  const int o = (blockIdx.x * blockDim.x + threadIdx.x) >> 5;  // wave per row
  const float* row = dec_W + (size_t)o * (2 * HIDDEN);
  float s = 0.0f;
  for (int t = lane; t < 2 * HIDDEN; t += 32) {
    float v = (t < HIDDEN) ? cnew_re[t] : cnew_im[t - HIDDEN];
    s = fmaf(row[t], v, s);
  }
  for (int off = 16; off > 0; off >>= 1) s += __shfl_down(s, off, 32);
  if (lane == 0) pred[o] = s + dec_b[o];
}

// ---------------- K6: lock + feedback + renorm + variance partials ------
__global__ void k6_final(const float2* __restrict__ newst,
                         const float* __restrict__ lasing,
                         const float* __restrict__ cnew_re,
                         const float* __restrict__ cnew_im,
                         const float* __restrict__ cphase,
                         float* __restrict__ tens) {
  __shared__ float s_amp[HIDDEN];
  __shared__ float s_red[256];
  const int i = blockIdx.x;
  const bool las = lasing[i] > 0.5f;     // block-uniform
  float lmax = 0.0f;
  for (int j = threadIdx.x; j < HIDDEN; j += blockDim.x) {
    float2 v = newst[(size_t)i * HIDDEN + j];
    if (las) {
      float a  = atan2f(v.y, v.x);
      float m  = __fsqrt_rn(v.x * v.x + v.y * v.y);
      float lk = 0.3f * cphase[j] + 0.7f * a;
      float sn, cs; __sincosf(lk, &sn, &cs);
      v.x = m * cs; v.y = m * sn;
    }
    v.x += 0.05f * cnew_re[j];
    v.y += 0.05f * cnew_im[j];
    float amp = __fsqrt_rn(v.x * v.x + v.y * v.y);
    s_amp[j] = amp;
    lmax = fmaxf(lmax, amp);
  }
  s_red[threadIdx.x] = lmax;
  __syncthreads();
  for (int off = blockDim.x >> 1; off > 0; off >>= 1) {
    if (threadIdx.x < off)
      s_red[threadIdx.x] = fmaxf(s_red[threadIdx.x], s_red[threadIdx.x + off]);
    __syncthreads();
  }
  const float inv = 1.0f / (s_red[0] + 1e-8f);
  float ls = 0.0f, lq = 0.0f;
  for (int j = threadIdx.x; j < HIDDEN; j += blockDim.x) {
    float a = s_amp[j] * inv;
    ls += a; lq = fmaf(a, a, lq);
  }
  __syncthreads();
  s_red[threadIdx.x] = ls; __syncthreads();
  for (int off = blockDim.x >> 1; off > 0; off >>= 1) {
    if (threadIdx.x < off) s_red[threadIdx.x] += s_red[threadIdx.x + off];
    __syncthreads();
  }
  const float bsum = s_red[0];
  __syncthreads();
  s_red[threadIdx.x] = lq; __syncthreads();
  for (int off = blockDim.x >> 1; off > 0; off >>= 1) {
    if (threadIdx.x < off) s_red[threadIdx.x] += s_red[threadIdx.x + off];
    __syncthreads();
  }
  if (threadIdx.x == 0) {
    atomicAdd(&tens[0], bsum);
    atomicAdd(&tens[1], s_red[0]);
  }
}

// ---------------- K7: tension = var(|final|) ----------------
__global__ void k7_tension(const float* __restrict__ tens,
                           float* __restrict__ out) {
  const float nT = (float)N_CELLS * (float)HIDDEN;
  float mean = tens[0] / nT;
  out[0] = tens[1] / nT - mean * mean;
}

// ---------------------------------------------------------------------------
extern "C" void kernel_launch(void* const* d_in, const int* in_sizes, int n_in,
                              void* d_out, int out_size, void* d_ws, size_t ws_size,
                              hipStream_t stream) {
  const float* x        = (const float*)d_in[0];
  const float* pump_W   = (const float*)d_in[1];
  const float* pump_b   = (const float*)d_in[2];
  const float* dec_W    = (const float*)d_in[3];
  const float* dec_b    = (const float*)d_in[4];
  const float* cs_re    = (const float*)d_in[5];
  const float* cs_im    = (const float*)d_in[6];
  const float* excited  = (const float*)d_in[7];
  const float* phase_vel= (const float*)d_in[8];
  const float* cav_re   = (const float*)d_in[9];
  const float* cav_im   = (const float*)d_in[10];
  const int*   neighbors= (const int*)d_in[11];
  float* out = (float*)d_out;

  // Workspace layout (bytes). Total ~134.3 MB.
  char* ws = (char*)d_ws;
  const size_t NH8 = (size_t)N_CELLS * HIDDEN * sizeof(float2);  // 67,108,864
  float2* rot    = (float2*)(ws);
  float2* newst  = (float2*)(ws + NH8);
  float*  lasing = (float*) (ws + 2 * NH8);                       // 64 KB
  float*  em     = (float*) (ws + 2 * NH8 + 65536);               // 1024 f
  int*    nl     = (int*)   (ws + 2 * NH8 + 65536 + 4096);
  float*  cnewr  = (float*) (ws + 2 * NH8 + 65536 + 4096 + 256);
  float*  cnewi  = cnewr + HIDDEN;
  float*  cphase = cnewi + HIDDEN;
  float*  tens   = cphase + HIDDEN;

  k0_init   <<<1, 1024, 0, stream>>>(em, nl, tens);
  k1_pump   <<<N_CELLS / 16 / 8, 256, 0, stream>>>(x, pump_W, pump_b, excited,
                                                   lasing, nl);
  k2_rotate <<<4096, 256, 0, stream>>>(cs_re, cs_im, phase_vel, rot);
  k3_couple <<<N_CELLS, 256, 0, stream>>>(rot, neighbors, newst);
  k4_emission<<<dim3(HIDDEN / 256, N_CELLS / 128), 256, 0, stream>>>(
      newst, lasing, em, em + HIDDEN);
  k5_cavity <<<1, HIDDEN, 0, stream>>>(em, em + HIDDEN, nl, cav_re, cav_im,
                                       cnewr, cnewi, cphase);
  k5b_pred  <<<DIM * 32 / 256, 256, 0, stream>>>(cnewr, cnewi, dec_W, dec_b, out);
  k6_final  <<<N_CELLS, 256, 0, stream>>>(newst, lasing, cnewr, cnewi, cphase,
                                          tens);
  k7_tension<<<1, 1, 0, stream>>>(tens, out + DIM);
}